// GravnetModel_50216757625458
// MI455X (gfx1250) — compile-verified
//
#include <hip/hip_runtime.h>
#include <hip/hip_bf16.h>
#include <math.h>

typedef __attribute__((ext_vector_type(2))) float v2f;
typedef __attribute__((ext_vector_type(8))) float v8f;
typedef __attribute__((ext_vector_type(4))) unsigned int v4u;
typedef __attribute__((ext_vector_type(8))) int v8i;
typedef __attribute__((ext_vector_type(4))) int v4i;

#define NPTS 8000
#define KMAX 128

#if defined(__HIP_DEVICE_COMPILE__)
#if __has_builtin(__builtin_amdgcn_tensor_load_to_lds) && __has_builtin(__builtin_amdgcn_s_wait_tensorcnt)
#define USE_TDM 1
#endif
#endif

__device__ __forceinline__ float elu1(float v) {
    return v > 0.0f ? v : (__expf(v) - 1.0f);
}

#ifdef USE_TDM
// Issue one TDM load of a [rows x K4] f32 tile (row stride ldw elements) from
// `gsrc` into LDS at `lds_off` (bytes).  Descriptor per CDNA5 ISA ch.8.
// Must be executed by exactly one wave of the workgroup.
__device__ __forceinline__ void tdm_load_tile(const float* gsrc, unsigned lds_off,
                                              int rows, int K4, int ldw) {
    unsigned long long ga = (unsigned long long)(const void*)gsrc;
    v4u g0;
    g0[0] = 1u;                                          // count=1, user mode
    g0[1] = lds_off;                                     // LDS byte address
    g0[2] = (unsigned)(ga & 0xFFFFFFFFu);                // global_addr[31:0]
    g0[3] = (unsigned)((ga >> 32) & 0x01FFFFFFu) | (2u << 30);  // addr[56:32] | type=2
    v8i g1;
    g1[0] = 0x00020000;                                  // data_size=2 (4 bytes)
    g1[1] = (int)((K4 & 0xFFFF) << 16);                  // tensor_dim0[15:0]
    g1[2] = (int)(((rows & 0xFFFF) << 16) |
                  ((unsigned)K4 >> 16));                 // tensor_dim0 hi | tensor_dim1 lo
    g1[3] = (int)((K4 & 0xFFFF) << 16);                  // tensor_dim1 hi=0 | tile_dim0
    g1[4] = rows & 0xFFFF;                               // tile_dim1 | tile_dim2=0
    g1[5] = ldw;                                         // tensor_dim0_stride[31:0]
    g1[6] = 0;                                           // stride hi | dim1_stride lo
    g1[7] = 0;                                           // tensor_dim1_stride hi
    v4i g2 = {0, 0, 0, 0};
    v4i g3 = {0, 0, 0, 0};
#if __clang_major__ >= 23
    v8i g4 = {0, 0, 0, 0, 0, 0, 0, 0};
    __builtin_amdgcn_tensor_load_to_lds(g0, g1, g2, g3, g4, 0);
#else
    __builtin_amdgcn_tensor_load_to_lds(g0, g1, g2, g3, 0);
#endif
}
#endif

// ---------------------------------------------------------------------------
// WMMA GEMM:  C[8000 x N] = act( A[8000 x K4] @ W[N x K4]^T + bias[N] )
// K4 compile-time multiple of 4; ragged operands zero-padded in workspace so
// all fragment loads are unconditional.  Weight panel [16 x K4] is staged
// into LDS by the Tensor Data Mover (one tensor_load_to_lds per block,
// s_wait_tensorcnt + barrier), then all 4 waves read B-fragments from LDS.
// Block = 128 threads (4 waves); wave w owns rows [64*bx + 16*w, +16).
// ---------------------------------------------------------------------------
template<int K4, bool ACT>
__global__ void gemm_wmma_kernel(const float* __restrict__ A, int lda,
                                 const float* __restrict__ W, int ldw,
                                 const float* __restrict__ bias,
                                 float* __restrict__ C, int ldc, int N) {
    const int wave = threadIdx.x >> 5;
    const int lane = threadIdx.x & 31;
    const int half = lane >> 4;
    const int l    = lane & 15;
    const int tile_m = (blockIdx.x * 4 + wave) * 16;
    const int tile_n = blockIdx.y * 16;
    const int row = tile_m + l;
    const int col = tile_n + l;

#ifdef USE_TDM
    __shared__ float wtile[16 * K4];
    if (threadIdx.x < 32) {
        tdm_load_tile(W + (size_t)tile_n * ldw,
                      (unsigned)(size_t)(&wtile[0]), 16, K4, ldw);
        __builtin_amdgcn_s_wait_tensorcnt(0);
    }
    __syncthreads();
    const float* bp = &wtile[(size_t)l * K4 + 2 * half];
#else
    const float* bp = W + (size_t)col * ldw + 2 * half;
#endif
    const float* ap = A + (size_t)row * lda + 2 * half;

    v8f acc = {};
#pragma unroll
    for (int k0 = 0; k0 < K4; k0 += 4) {
        v2f a = *(const v2f*)(ap + k0);
        v2f b = *(const v2f*)(bp + k0);
        acc = __builtin_amdgcn_wmma_f32_16x16x4_f32(
            false, a, false, b, (short)0, acc, false, false);
    }

    if (col < N) {
        const float bv = bias ? bias[col] : 0.0f;
#pragma unroll
        for (int v = 0; v < 8; ++v) {
            float val = acc[v] + bv;
            if (ACT) val = elu1(val);
            C[(size_t)(tile_m + v + 8 * half) * ldc + col] = val;
        }
    }
}

template<int K4>
static void launch_gemm(const float* A, int lda, const float* W, int ldw,
                        const float* bias, float* C, int ldc, int N, int act,
                        hipStream_t stream) {
    dim3 g(NPTS / 64, (N + 15) / 16);
    if (act)
        gemm_wmma_kernel<K4, true><<<g, 128, 0, stream>>>(A, lda, W, ldw, bias, C, ldc, N);
    else
        gemm_wmma_kernel<K4, false><<<g, 128, 0, stream>>>(A, lda, W, ldw, bias, C, ldc, N);
}

// Zero-padded weight packing: dst[N16 x K4] <- src[N x K]
__global__ void pack_w_kernel(float* __restrict__ dst, int N16, int K4,
                              const float* __restrict__ src, int N, int K) {
    int t = blockIdx.x * blockDim.x + threadIdx.x;
    if (t >= N16 * K4) return;
    int r = t / K4, c = t % K4;
    dst[t] = (r < N && c < K) ? src[r * K + c] : 0.0f;
}

// ---------------------------------------------------------------------------
// Column mean/var (population) for batchnorm.  One block per column.
// ---------------------------------------------------------------------------
__global__ void col_stats_kernel(const float* __restrict__ X, int M, int N, int ld,
                                 float* __restrict__ stats) {
    __shared__ float sh1[256];
    __shared__ float sh2[256];
    const int c = blockIdx.x;
    float s = 0.0f, s2 = 0.0f;
    for (int r = threadIdx.x; r < M; r += blockDim.x) {
        float v = X[(size_t)r * ld + c];
        s += v; s2 += v * v;
    }
    sh1[threadIdx.x] = s; sh2[threadIdx.x] = s2;
    __syncthreads();
    for (int off = 128; off > 0; off >>= 1) {
        if (threadIdx.x < off) {
            sh1[threadIdx.x] += sh1[threadIdx.x + off];
            sh2[threadIdx.x] += sh2[threadIdx.x + off];
        }
        __syncthreads();
    }
    if (threadIdx.x == 0) {
        float m = sh1[0] / (float)M;
        stats[c]     = m;
        stats[N + c] = sh2[0] / (float)M - m * m;
    }
}

__global__ void bn_apply_kernel(const float* __restrict__ X, int ldx,
                                float* __restrict__ Y, int ldy,
                                const float* __restrict__ stats,
                                const float* __restrict__ g,
                                const float* __restrict__ b,
                                int M, int N) {
    int t = blockIdx.x * blockDim.x + threadIdx.x;
    if (t >= M * N) return;
    int r = t / N, c = t % N;
    float m = stats[c], v = stats[N + c];
    Y[(size_t)r * ldy + c] =
        (X[(size_t)r * ldx + c] - m) * rsqrtf(v + 1e-5f) * g[c] + b[c];
}

__global__ void copy_cols_kernel(float* __restrict__ dst, int ldd, int doff,
                                 const float* __restrict__ src, int lds, int soff,
                                 int ncols, int M) {
    int t = blockIdx.x * blockDim.x + threadIdx.x;
    if (t >= M * ncols) return;
    int r = t / ncols, c = t % ncols;
    dst[(size_t)r * ldd + doff + c] = src[(size_t)r * lds + soff + c];
}

__global__ void fill_zero_kernel(float* __restrict__ p, int count) {
    int t = blockIdx.x * blockDim.x + threadIdx.x;
    if (t < count) p[t] = 0.0f;
}

// c = s + 1000*batch, padded to 4; sq = |c|^2.  Writes zero padding for
// i in [M, Mpad) so the kNN candidate loop can load unconditionally.
__global__ void prep_coords_kernel(const float* __restrict__ s, int lds,
                                   const int* __restrict__ batch,
                                   float* __restrict__ c4, float* __restrict__ sq,
                                   int M, int Mpad) {
    int i = blockIdx.x * blockDim.x + threadIdx.x;
    if (i >= Mpad) return;
    float x = 0.0f, y = 0.0f, z = 0.0f;
    if (i < M) {
        float off = 1000.0f * (float)batch[i];
        x = s[(size_t)i * lds + 0] + off;
        y = s[(size_t)i * lds + 1] + off;
        z = s[(size_t)i * lds + 2] + off;
    }
    c4[i * 4 + 0] = x; c4[i * 4 + 1] = y; c4[i * 4 + 2] = z; c4[i * 4 + 3] = 0.0f;
    sq[i] = x * x + y * y + z * z;
}

// ---------------------------------------------------------------------------
// Fused kNN: block of 256 threads (8 waves) owns 16 query rows.
// Wave w computes a 16x16 Gram tile with one V_WMMA_F32_16X16X4_F32 against
// candidates [cb+16w, cb+16w+16); 16 threads merge the 16x128 distance tile
// into per-row sorted top-K lists in LDS.  Candidate loads are unconditional
// (buffers padded past M); out-of-range handled by a single select on |c|^2.
// ---------------------------------------------------------------------------
__global__ void knn_topk_kernel(const float* __restrict__ c4,
                                const float* __restrict__ sq,
                                int M, int K,
                                int* __restrict__ oidx, float* __restrict__ od2) {
    __shared__ float dtile[16 * 128];
    __shared__ float bd[16 * KMAX];
    __shared__ int   bi[16 * KMAX];
    __shared__ float shq[16];

    const int qbase = blockIdx.x * 16;
    const int tid  = threadIdx.x;
    const int wave = tid >> 5;
    const int lane = tid & 31;
    const int half = lane >> 4;
    const int l    = lane & 15;
    const int ka   = 2 * half;

    for (int i = tid; i < 16 * K; i += blockDim.x) { bd[i] = 3.0e38f; bi[i] = 0; }
    if (tid < 16) shq[tid] = sq[qbase + tid];   // M % 16 == 0
    __syncthreads();

    const int qrow = qbase + l;
    const v2f a = *(const v2f*)(c4 + (size_t)qrow * 4 + ka);

    for (int cb = 0; cb < M; cb += 128) {
        const int cand = cb + wave * 16 + l;          // < M + 127 (padded)
        const v2f b = *(const v2f*)(c4 + (size_t)cand * 4 + ka);
        const float sqv = sq[cand];
        const float sqc = (cand < M) ? sqv : 3.0e38f;

        {   // prefetch next candidate strip (clamped: always a valid address)
            const int pre = (cb + 128 < M) ? (cb + 128 + wave * 16 + l) : 0;
            __builtin_prefetch(c4 + (size_t)pre * 4, 0, 1);
        }

        v8f acc = {};
        acc = __builtin_amdgcn_wmma_f32_16x16x4_f32(
            false, a, false, b, (short)0, acc, false, false);

#pragma unroll
        for (int v = 0; v < 8; ++v) {
            const int r = v + 8 * half;
            float d = shq[r] + sqc - 2.0f * acc[v];
            if (qbase + r == cand) d += 1.0e9f;   // self mask (matches +eye*1e9)
            dtile[r * 128 + wave * 16 + l] = d;
        }
        __syncthreads();

        if (tid < 16) {
            const int row = tid;
            float* rbd = &bd[row * K];
            int*   rbi = &bi[row * K];
            const int lim = (M - cb < 128) ? (M - cb) : 128;
            for (int c = 0; c < lim; ++c) {
                float d = dtile[row * 128 + c];
                if (d < rbd[K - 1]) {
                    int j = K - 1;
                    while (j > 0 && rbd[j - 1] > d) {
                        rbd[j] = rbd[j - 1]; rbi[j] = rbi[j - 1]; --j;
                    }
                    rbd[j] = d; rbi[j] = cb + c;
                }
            }
        }
        __syncthreads();
    }

    for (int i = tid; i < 16 * K; i += blockDim.x) {
        const int r = i / K, c = i % K;
        const int grow = qbase + r;
        oidx[(size_t)grow * K + c] = bi[i];
        od2[(size_t)grow * K + c]  = fmaxf(bd[i], 0.0f);
    }
}

// msg = feat[idx] * exp(-10*d2); mean & max over K neighbors -> hcat[:,32:96]
__global__ void aggregate_kernel(const float* __restrict__ feat,
                                 const int* __restrict__ idx,
                                 const float* __restrict__ d2,
                                 int K, float* __restrict__ hcat, int M) {
    const int i = blockIdx.x;
    const int f = threadIdx.x;  // 0..31
    float sum = 0.0f, mx = -3.0e38f;
    for (int n = 0; n < K; ++n) {
        const int j = idx[(size_t)i * K + n];
        const float w = __expf(-10.0f * d2[(size_t)i * K + n]);
        const float v = feat[(size_t)j * 32 + f] * w;
        sum += v;
        mx = fmaxf(mx, v);
    }
    hcat[(size_t)i * 96 + 32 + f] = sum / (float)K;
    hcat[(size_t)i * 96 + 64 + f] = mx;
}

__global__ void norm_out_kernel(const float* __restrict__ ob,
                                float* __restrict__ out, int M) {
    int i = blockIdx.x * blockDim.x + threadIdx.x;
    if (i >= M) return;
    float x = ob[i * 4 + 0], y = ob[i * 4 + 1], z = ob[i * 4 + 2], w = ob[i * 4 + 3];
    float n = fmaxf(sqrtf(x * x + y * y + z * z), 1e-12f);
    out[i * 4 + 0] = x / n;
    out[i * 4 + 1] = y / n;
    out[i * 4 + 2] = z / n;
    out[i * 4 + 3] = w;
}

// ---------------------------------------------------------------------------
extern "C" void kernel_launch(void* const* d_in, const int* in_sizes, int n_in,
                              void* d_out, int out_size, void* d_ws, size_t ws_size,
                              hipStream_t stream) {
    const int M = NPTS;
    const int MP = NPTS + 128;   // padded for branch-free kNN loads
    const float* x      = (const float*)d_in[0];
    const int*   batch  = (const int*)  d_in[1];
    const float* bn0_g  = (const float*)d_in[2];
    const float* bn0_b  = (const float*)d_in[3];
    const float* Wd1    = (const float*)d_in[4];   // 64x8
    const float* preW1  = (const float*)d_in[5];   // 4x32x64
    const float* preb1  = (const float*)d_in[6];
    const float* preW2  = (const float*)d_in[7];   // 4x32x32
    const float* preb2  = (const float*)d_in[8];
    const float* bn1g   = (const float*)d_in[9];
    const float* bn1b   = (const float*)d_in[10];
    const float* Ws     = (const float*)d_in[11];  // 4x3x32
    const float* bs     = (const float*)d_in[12];
    const float* Wh     = (const float*)d_in[13];  // 4x32x32
    const float* bh     = (const float*)d_in[14];
    const float* Wo     = (const float*)d_in[15];  // 4x32x96
    const float* bo     = (const float*)d_in[16];
    const float* postW1 = (const float*)d_in[17];  // 4x32x67
    const float* postb1 = (const float*)d_in[18];
    const float* postW2 = (const float*)d_in[19];  // 4x32x32
    const float* postb2 = (const float*)d_in[20];
    const float* bn2g   = (const float*)d_in[21];
    const float* bn2b   = (const float*)d_in[22];
    const float* pgW0   = (const float*)d_in[23];  // 64x192
    const float* pgb0   = (const float*)d_in[24];
    const float* pgW    = (const float*)d_in[25];  // 2x64x64
    const float* pgb    = (const float*)d_in[26];
    const float* Wout   = (const float*)d_in[27];  // 4x64
    const float* bout   = (const float*)d_in[28];

    // ---- workspace carve (floats) ----
    float* base = (float*)d_ws;
    size_t o = 0;
    auto carve = [&](size_t n) { float* p = base + o; o += n; return p; };
    float* xin   = carve((size_t)M * 64);
    float* bnx   = carve((size_t)M * 8);
    float* h     = carve((size_t)M * 32);
    float* h2    = carve((size_t)M * 32);
    float* tmp   = carve((size_t)M * 32);
    float* sbuf  = carve((size_t)M * 4);
    float* c4    = carve((size_t)MP * 4);
    float* sqb   = carve((size_t)MP);
    float* feat  = carve((size_t)M * 32);
    float* hcat  = carve((size_t)M * 96);
    float* xcat  = carve((size_t)M * 68);
    float* xgn   = carve((size_t)M * 32);
    float* xc    = carve((size_t)M * 32);
    float* allf  = carve((size_t)M * 192);
    float* f1    = carve((size_t)M * 64);
    float* f2    = carve((size_t)M * 64);
    float* ob    = carve((size_t)M * 4);
    float* stats = carve(256);
    float* wsp   = carve(4 * 16 * 32);   // packed Ws     -> [16 x 32] per layer
    float* wop1p = carve(4 * 32 * 68);   // packed postW1 -> [32 x 68] per layer
    float* woutp = carve(16 * 64);       // packed Wout   -> [16 x 64]
    float* d2b   = carve((size_t)M * KMAX);
    int*   idxb  = (int*)carve((size_t)M * KMAX);

    const int klist[4] = {16, 128, 16, 128};
    auto ew = [&](int count) { return dim3((count + 255) / 256); };

    // ---- pack ragged weights into zero-padded tiles; zero xcat pad column ----
    for (int i = 0; i < 4; ++i) {
        pack_w_kernel<<<ew(16 * 32), 256, 0, stream>>>(wsp + i * 16 * 32, 16, 32,
                                                       Ws + i * 3 * 32, 3, 32);
        pack_w_kernel<<<ew(32 * 68), 256, 0, stream>>>(wop1p + i * 32 * 68, 32, 68,
                                                       postW1 + i * 32 * 67, 32, 67);
    }
    pack_w_kernel<<<ew(16 * 64), 256, 0, stream>>>(woutp, 16, 64, Wout, 4, 64);
    fill_zero_kernel<<<ew(M * 68), 256, 0, stream>>>(xcat, M * 68);

    // ---- input BN + embedding ----
    col_stats_kernel<<<8, 256, 0, stream>>>(x, M, 8, 8, stats);
    bn_apply_kernel<<<ew(M * 8), 256, 0, stream>>>(x, 8, bnx, 8, stats, bn0_g, bn0_b, M, 8);
    launch_gemm<8>(bnx, 8, Wd1, 8, nullptr, xin, 64, 64, 0, stream);
    copy_cols_kernel<<<ew(M * 64), 256, 0, stream>>>(allf, 192, 0, xin, 64, 0, 64, M);

    for (int i = 0; i < 4; ++i) {
        const int K = klist[i];
        if (i > 0) {
            fill_zero_kernel<<<ew(M * 64), 256, 0, stream>>>(xin, M * 64);
            copy_cols_kernel<<<ew(M * 32), 256, 0, stream>>>(xin, 64, 0, xc, 32, 0, 32, M);
        }
        // pre-MLP
        launch_gemm<64>(xin, 64, preW1 + i * 32 * 64, 64, preb1 + i * 32, h, 32, 32, 1, stream);
        launch_gemm<32>(h, 32, preW2 + i * 32 * 32, 32, preb2 + i * 32, h2, 32, 32, 1, stream);
        col_stats_kernel<<<32, 256, 0, stream>>>(h2, M, 32, 32, stats);
        bn_apply_kernel<<<ew(M * 32), 256, 0, stream>>>(h2, 32, h, 32, stats,
                                                        bn1g + i * 32, bn1b + i * 32, M, 32);
        // learned space + features
        launch_gemm<32>(h, 32, wsp + i * 16 * 32, 32, bs + i * 3, sbuf, 4, 3, 0, stream);
        launch_gemm<32>(h, 32, Wh + i * 32 * 32, 32, bh + i * 32, feat, 32, 32, 0, stream);
        // kNN
        prep_coords_kernel<<<ew(MP), 256, 0, stream>>>(sbuf, 4, batch, c4, sqb, M, MP);
        knn_topk_kernel<<<M / 16, 256, 0, stream>>>(c4, sqb, M, K, idxb, d2b);
        // aggregate -> [h | mean | max]
        copy_cols_kernel<<<ew(M * 32), 256, 0, stream>>>(hcat, 96, 0, h, 32, 0, 32, M);
        aggregate_kernel<<<M, 32, 0, stream>>>(feat, idxb, d2b, K, hcat, M);
        launch_gemm<96>(hcat, 96, Wo + i * 32 * 96, 96, bo + i * 32, xgn, 32, 32, 0, stream);
        // post: [xgn | s | h]  (col 67 pre-zeroed)
        copy_cols_kernel<<<ew(M * 32), 256, 0, stream>>>(xcat, 68, 0, xgn, 32, 0, 32, M);
        copy_cols_kernel<<<ew(M * 3), 256, 0, stream>>>(xcat, 68, 32, sbuf, 4, 0, 3, M);
        copy_cols_kernel<<<ew(M * 32), 256, 0, stream>>>(xcat, 68, 35, h, 32, 0, 32, M);
        launch_gemm<68>(xcat, 68, wop1p + i * 32 * 68, 68, postb1 + i * 32, h2, 32, 32, 1, stream);
        launch_gemm<32>(h2, 32, postW2 + i * 32 * 32, 32, postb2 + i * 32, tmp, 32, 32, 1, stream);
        col_stats_kernel<<<32, 256, 0, stream>>>(tmp, M, 32, 32, stats);
        bn_apply_kernel<<<ew(M * 32), 256, 0, stream>>>(tmp, 32, xc, 32, stats,
                                                        bn2g + i * 32, bn2b + i * 32, M, 32);
        copy_cols_kernel<<<ew(M * 32), 256, 0, stream>>>(allf, 192, 64 + 32 * i, xc, 32, 0, 32, M);
    }

    // ---- head ----
    launch_gemm<192>(allf, 192, pgW0, 192, pgb0, f1, 64, 64, 1, stream);
    launch_gemm<64>(f1, 64, pgW, 64, pgb, f2, 64, 64, 1, stream);
    launch_gemm<64>(f2, 64, pgW + 64 * 64, 64, pgb + 64, f1, 64, 64, 1, stream);
    launch_gemm<64>(f1, 64, woutp, 64, bout, ob, 4, 4, 0, stream);
    norm_out_kernel<<<ew(M), 256, 0, stream>>>(ob, (float*)d_out, M);
    (void)in_sizes; (void)n_in; (void)out_size; (void)ws_size;
}